// decode_param_CNN_Fourier_83829171683376
// MI455X (gfx1250) — compile-verified
//
#include <hip/hip_runtime.h>
#include <cmath>

typedef __attribute__((ext_vector_type(2))) float v2f;
typedef __attribute__((ext_vector_type(8))) float v8f;

#define N_T      48
#define NF       6
#define NPARAMS  2688
#define M_ROWS   288      // N_T * NF
#define MT       18       // 288 / 16
#define NP       16       // pixels per block (one WMMA N-tile)
#define PDIM     10000

#define SA_STRIDE 9       // A[288][8] padded
#define SB_STRIDE 17      // B[8][16]  padded
#define SF_STRIDE 17      // fields staging [288][16] padded

__device__ __forceinline__ float softplus_f(float x) {
    // stable softplus: max(x,0) + log(1 + exp(-|x|))
    float e = __expf(-fabsf(x));
    return fmaxf(x, 0.0f) + __logf(1.0f + e);
}

__global__ __launch_bounds__(256) void fourier_decode_kernel(
    const float* __restrict__ params, float* __restrict__ out)
{
    __shared__ float sA[M_ROWS * SA_STRIDE];  // coeff matrix A[(t*6+f)][(i*2+c)]
    __shared__ float sB[8 * SB_STRIDE];       // basis    B[(i*2+c)][p_local]
    __shared__ float sF[M_ROWS * SF_STRIDE];  // fields   F[(t*6+f)][p_local]

    const int tid = threadIdx.x;
    const int b   = blockIdx.y;
    const int p0  = blockIdx.x * NP;
    const float* pb = params + b * NPARAMS;

    // ---- Phase A: load coefficient matrix A[288][8] into LDS ----
    // coeffs[b,t,i,f,c] = params[b, start(t,i) + 2f + c]
    // start(t,0)=8t+8, start(t,1)=8, start(t,2)=40, start(t,3)=52
    for (int flat = tid; flat < M_ROWS * 8; flat += 256) {
        int Mr = flat >> 3;
        int k  = flat & 7;          // k = i*2 + c
        int t = Mr / 6, f = Mr - 6 * t;
        int i = k >> 1, c = k & 1;
        int start = (i == 0) ? (8 * t + 8) : ((i == 1) ? 8 : ((i == 2) ? 40 : 52));
        sA[Mr * SA_STRIDE + k] = pb[start + 2 * f + c];
    }

    // ---- Phase B: basis B[8][16] = {cos,sin}(2pi(k*gx + l*gy)/100) ----
    if (tid < NP * 8) {
        int p_local = tid >> 3;
        int k = tid & 7;
        int i = k >> 1, c = k & 1;
        float kx = (i == 0) ? 0.0f : 1.0f;
        float ly = (i == 1) ? -1.0f : ((i == 2) ? 0.0f : 1.0f);
        int pg = p0 + p_local;
        float gx = (float)(pg / 100);
        float gy = (float)(pg % 100);
        const float W = 0.06283185307179586477f;  // 2*pi/100
        float ph = W * (kx * gx + ly * gy);
        float s, co;
        __sincosf(ph, &s, &co);
        sB[k * SB_STRIDE + p_local] = c ? s : co;
    }

    __syncthreads();

    // ---- Phase C: fields[288][16] = A[288][8] x B[8][16] via WMMA f32 16x16x4 ----
    const int lane  = tid & 31;
    const int wave  = tid >> 5;
    const int mrow  = lane & 15;          // M within tile (A) / N col (B,D)
    const int khalf = (lane >> 4) << 1;   // lanes 0-15 -> K{0,1}; 16-31 -> K{2,3}
    for (int m = wave; m < MT; m += 8) {
        int row = m * 16 + mrow;
        const float* arow = &sA[row * SA_STRIDE];
        v2f a0, a1, b0, b1;
        a0.x = arow[khalf + 0];
        a0.y = arow[khalf + 1];
        a1.x = arow[4 + khalf + 0];
        a1.y = arow[4 + khalf + 1];
        b0.x = sB[(khalf + 0) * SB_STRIDE + mrow];
        b0.y = sB[(khalf + 1) * SB_STRIDE + mrow];
        b1.x = sB[(4 + khalf + 0) * SB_STRIDE + mrow];
        b1.y = sB[(4 + khalf + 1) * SB_STRIDE + mrow];
        v8f acc = {};
        acc = __builtin_amdgcn_wmma_f32_16x16x4_f32(false, a0, false, b0,
                                                    (short)0, acc, false, false);
        acc = __builtin_amdgcn_wmma_f32_16x16x4_f32(false, a1, false, b1,
                                                    (short)0, acc, false, false);
        // D tile: VGPR r, lane L -> (M = m*16 + r + 8*(L/16), N = L&15)
        int rbase = m * 16 + 8 * (lane >> 4);
        #pragma unroll
        for (int r = 0; r < 8; ++r)
            sF[(rbase + r) * SF_STRIDE + mrow] = acc[r];
    }

    __syncthreads();

    // ---- Phase D: epilogue + coalesced transposed stores ----
    float gamma = softplus_f(pb[44]);
    float beta  = softplus_f(pb[45]);
    float bias0 = pb[46], bias1 = pb[47], bias2 = pb[48];
    float bias3 = pb[49], bias4 = pb[50], bias5 = pb[51];

    const int O_M   = 7680000;    // kappa: 16*480000
    const int O_H   = 23040000;   // + m: 16*2*480000
    const int O_TAU = 53760000;   // + H: 16*4*480000

    for (int q = tid; q < NP * N_T; q += 256) {
        int p_local = q / N_T;
        int t = q - p_local * N_T;
        const float* frow = &sF[(t * 6) * SF_STRIDE + p_local];
        float f0 = frow[0 * SF_STRIDE] + bias0;
        float f1 = frow[1 * SF_STRIDE] + bias1;
        float f2 = frow[2 * SF_STRIDE] + bias2;
        float f3 = frow[3 * SF_STRIDE] + bias3;
        float f4 = frow[4 * SF_STRIDE] + bias4;
        float f5 = frow[5 * SF_STRIDE] + bias5;

        float kappa = softplus_f(f0);
        float tau   = softplus_f(f5);
        float hxy   = beta * f3 * f4;

        int pt = (p0 + p_local) * 48 + t;   // contiguous across q -> coalesced
        out[b * 480000 + pt]                    = kappa;
        out[O_M + (b * 2 + 0) * 480000 + pt]    = f1;
        out[O_M + (b * 2 + 1) * 480000 + pt]    = f2;
        out[O_H + (b * 4 + 0) * 480000 + pt]    = gamma + beta * f3 * f3;
        out[O_H + (b * 4 + 1) * 480000 + pt]    = hxy;
        out[O_H + (b * 4 + 2) * 480000 + pt]    = hxy;
        out[O_H + (b * 4 + 3) * 480000 + pt]    = gamma + beta * f4 * f4;
        out[O_TAU + b * 480000 + pt]            = tau;
    }
}

extern "C" void kernel_launch(void* const* d_in, const int* in_sizes, int n_in,
                              void* d_out, int out_size, void* d_ws, size_t ws_size,
                              hipStream_t stream) {
    const float* params = (const float*)d_in[0];
    float* out = (float*)d_out;
    dim3 grid(PDIM / NP, 16);   // 625 pixel tiles x 16 batches
    fourier_decode_kernel<<<grid, 256, 0, stream>>>(params, out);
}